// ModifiedSingleStreamBlock_77781857730929
// MI455X (gfx1250) — compile-verified
//
#include <hip/hip_runtime.h>
#include <cstddef>
#include <cstdint>

typedef __attribute__((ext_vector_type(2))) float v2f;
typedef __attribute__((ext_vector_type(8))) float v8f;

#define HID   1536
#define QKVW  4608
#define MLPW  6144
#define CATW  7680
#define HEADS 12
#define HD    128
#define LXN   4352
#define LCN   4116
#define NCON  20

__device__ __forceinline__ v8f wmma4(v2f a, v2f b, v8f c) {
  // V_WMMA_F32_16X16X4_F32 : D = A(16x4,f32) * B(4x16,f32) + C(16x16,f32)
  return __builtin_amdgcn_wmma_f32_16x16x4_f32(false, a, false, b, (short)0, c,
                                               false, false);
}

// Direct global -> LDS async copy (16B per lane), tracked with ASYNCcnt.
__device__ __forceinline__ void async_b128(unsigned lds, const void* gaddr) {
  asm volatile("global_load_async_to_lds_b128 %0, %1, off"
               :
               : "v"(lds), "v"((unsigned long long)(uintptr_t)gaddr)
               : "memory");
}
__device__ __forceinline__ void wait_async0() {
  asm volatile("s_wait_asynccnt 0x0" ::: "memory");
}

// ---------------------------------------------------------------------------
// tiny elementwise / vector kernels
// ---------------------------------------------------------------------------
__global__ __launch_bounds__(256) void silu_k(const float* __restrict__ vec,
                                              float* __restrict__ sv) {
  int i = blockIdx.x * 256 + threadIdx.x;
  if (i < HID) {
    float x = vec[i];
    sv[i] = x / (1.0f + __expf(-x));
  }
}

// m = silu(vec) @ mod_w^T + mod_b   (4608 x 1536 mat-vec, tiny)
__global__ __launch_bounds__(256) void modm_k(const float* __restrict__ sv,
                                              const float* __restrict__ mw,
                                              const float* __restrict__ mb,
                                              float* __restrict__ m) {
  int o = blockIdx.x * 256 + threadIdx.x;
  if (o >= 3 * HID) return;
  const float* w = mw + (size_t)o * HID;
  float acc = 0.f;
  for (int k = 0; k < HID; ++k) acc += sv[k] * w[k];
  m[o] = acc + mb[o];
}

// xm[row] = (1+scale)*layernorm(src[row]) + shift ; src = concat(srcA[:nA], srcB)
__global__ __launch_bounds__(256) void ln_mod_k(const float* __restrict__ srcA,
                                                int nA,
                                                const float* __restrict__ srcB,
                                                const float* __restrict__ mvec,
                                                float* __restrict__ xm) {
  int row = blockIdx.x;
  const float* src = (row < nA) ? (srcA + (size_t)row * HID)
                                : (srcB + (size_t)(row - nA) * HID);
  int tid = threadIdx.x;
  float v[6];
  float s = 0.f, ss = 0.f;
#pragma unroll
  for (int j = 0; j < 6; ++j) {
    v[j] = src[tid + j * 256];
    s += v[j];
    ss += v[j] * v[j];
  }
#pragma unroll
  for (int off = 16; off; off >>= 1) {
    s += __shfl_xor(s, off);
    ss += __shfl_xor(ss, off);
  }
  __shared__ float redS[8], redQ[8];
  int lane = tid & 31, wv = tid >> 5;
  if (lane == 0) { redS[wv] = s; redQ[wv] = ss; }
  __syncthreads();
  float ts = 0.f, tq = 0.f;
#pragma unroll
  for (int w = 0; w < 8; ++w) { ts += redS[w]; tq += redQ[w]; }
  float mu = ts * (1.0f / HID);
  float var = tq * (1.0f / HID) - mu * mu;
  float inv = rsqrtf(var + 1e-6f);
  float* dst = xm + (size_t)row * HID;
#pragma unroll
  for (int j = 0; j < 6; ++j) {
    int c = tid + j * 256;
    float ln = (v[j] - mu) * inv;
    dst[c] = (1.0f + mvec[HID + c]) * ln + mvec[c];
  }
}

// rmsnorm(q,k) + rope + q-prescale ; qkv layout [L, 4608] (q|k|v)
__global__ __launch_bounds__(256) void rope_rms_k(
    const float* __restrict__ qkv, const float* __restrict__ pe,
    const float* __restrict__ qs, const float* __restrict__ ks,
    float* __restrict__ Qr, float* __restrict__ Kr, int L) {
  int lane = threadIdx.x & 31, wave = threadIdx.x >> 5;
  int g = blockIdx.x * 8 + wave;
  if (g >= L * HEADS) return;
  int row = g / HEADS, h = g - row * HEADS;
  int d0 = lane * 4;
  const float* base = qkv + (size_t)row * QKVW + h * HD + d0;
  float q[4], k[4];
#pragma unroll
  for (int i = 0; i < 4; ++i) { q[i] = base[i]; k[i] = base[HID + i]; }
  float sq = 0.f, sk = 0.f;
#pragma unroll
  for (int i = 0; i < 4; ++i) { sq += q[i] * q[i]; sk += k[i] * k[i]; }
#pragma unroll
  for (int off = 16; off; off >>= 1) {
    sq += __shfl_xor(sq, off);
    sk += __shfl_xor(sk, off);
  }
  float rq = rsqrtf(sq * (1.0f / HD) + 1e-6f);
  float rk = rsqrtf(sk * (1.0f / HD) + 1e-6f);
#pragma unroll
  for (int i = 0; i < 4; ++i) {
    q[i] *= rq * qs[d0 + i];
    k[i] *= rk * ks[d0 + i];
  }
  float qo[4], ko[4];
#pragma unroll
  for (int p = 0; p < 2; ++p) {
    int i = lane * 2 + p;
    const float* pp = pe + (((size_t)row * 64 + i) * 4);
    float a = pp[0], b = pp[1], c2 = pp[2], d2 = pp[3];
    qo[2 * p]     = a * q[2 * p] + b * q[2 * p + 1];
    qo[2 * p + 1] = c2 * q[2 * p] + d2 * q[2 * p + 1];
    ko[2 * p]     = a * k[2 * p] + b * k[2 * p + 1];
    ko[2 * p + 1] = c2 * k[2 * p] + d2 * k[2 * p + 1];
  }
  const float scale = 0.08838834764831845f;  // 1/sqrt(128), folded into Q
  float* qd = Qr + (size_t)row * HID + h * HD + d0;
  float* kd = Kr + (size_t)row * HID + h * HD + d0;
#pragma unroll
  for (int i = 0; i < 4; ++i) { qd[i] = qo[i] * scale; kd[i] = ko[i]; }
}

// ---------------------------------------------------------------------------
// WMMA f32 GEMM:  C[M,N] = epi(A[M,K] @ B[N,K]^T + bias)
// 128x128 block tile, 8 waves (2x4), wave tile 64x32, K-step 16.
// Tiles staged by GLOBAL_LOAD_ASYNC_TO_LDS_B128 with double buffering.
// Row stride 20 floats (80B): 16B-aligned rows + conflict-free frag reads.
// epi: 0 = none, 1 = tanh-GELU, 2 = res + gate*val
// ---------------------------------------------------------------------------
#define BM 128
#define BN 128
#define BK 16
#define BKP 20
#define TILEB (BM * BKP * 4)  // bytes per LDS tile buffer

__global__ __launch_bounds__(256) void gemm_wmma_f32(
    const float* __restrict__ A, int lda, const float* __restrict__ B, int ldb,
    const float* __restrict__ bias, float* __restrict__ C, int ldc, int M,
    int N, int K, int epi, const float* __restrict__ res, int ldres,
    const float* __restrict__ gate) {
  __shared__ float As[2][BM * BKP];
  __shared__ float Bs[2][BM * BKP];
  int tid = threadIdx.x;
  int lane = tid & 31, wave = tid >> 5;
  int wm = wave >> 2, wn = wave & 3;  // waves 2 x 4
  int fr = lane & 15, hi = lane >> 4;
  int m0 = blockIdx.y * BM, n0 = blockIdx.x * BN;

  v8f acc[4][2];
#pragma unroll
  for (int i = 0; i < 4; ++i)
#pragma unroll
    for (int j = 0; j < 2; ++j) acc[i][j] = (v8f){};

  int lrow = tid >> 1;         // 0..127
  int lcol = (tid & 1) * 8;    // 0 or 8
  int gar = min(m0 + lrow, M - 1);
  int gbr = min(n0 + lrow, N - 1);
  const unsigned asB = (unsigned)(uintptr_t)(void*)As;
  const unsigned bsB = (unsigned)(uintptr_t)(void*)Bs;
  const unsigned ldsOff = (unsigned)(lrow * BKP * 4 + lcol * 4);
  const float* gA = A + (size_t)gar * lda + lcol;
  const float* gB = B + (size_t)gbr * ldb + lcol;

  auto issue = [&](int buf, int k0) {
    unsigned la = asB + buf * TILEB + ldsOff;
    unsigned lb = bsB + buf * TILEB + ldsOff;
    const char* ga = (const char*)(gA + k0);
    const char* gb = (const char*)(gB + k0);
    async_b128(la, ga);
    async_b128(la + 16u, ga + 16);
    async_b128(lb, gb);
    async_b128(lb + 16u, gb + 16);
  };

  int nk = K / BK;
  issue(0, 0);
  for (int it = 0; it < nk; ++it) {
    wait_async0();      // my async copies into buf it&1 are done
    __syncthreads();    // everyone's are; prior reads of other buf done too
    if (it + 1 < nk) issue((it + 1) & 1, (it + 1) * BK);
    const float* as = As[it & 1];
    const float* bs = Bs[it & 1];
#pragma unroll
    for (int kk = 0; kk < BK; kk += 4) {
      int fc = kk + 2 * hi;
      v2f af[4], bf[2];
#pragma unroll
      for (int mt = 0; mt < 4; ++mt) {
        float2 t = *(const float2*)(as + (wm * 64 + mt * 16 + fr) * BKP + fc);
        af[mt].x = t.x;
        af[mt].y = t.y;
      }
#pragma unroll
      for (int nt = 0; nt < 2; ++nt) {
        float2 t = *(const float2*)(bs + (wn * 32 + nt * 16 + fr) * BKP + fc);
        bf[nt].x = t.x;
        bf[nt].y = t.y;
      }
#pragma unroll
      for (int mt = 0; mt < 4; ++mt)
#pragma unroll
        for (int nt = 0; nt < 2; ++nt)
          acc[mt][nt] = wmma4(af[mt], bf[nt], acc[mt][nt]);
    }
  }
  // epilogue
#pragma unroll
  for (int mt = 0; mt < 4; ++mt)
#pragma unroll
    for (int nt = 0; nt < 2; ++nt) {
      int colg = n0 + wn * 32 + nt * 16 + fr;
#pragma unroll
      for (int r = 0; r < 8; ++r) {
        int rowg = m0 + wm * 64 + mt * 16 + r + hi * 8;
        if (rowg < M && colg < N) {
          float v = acc[mt][nt][r] + bias[colg];
          if (epi == 1) {
            float u = 0.7978845608028654f * (v + 0.044715f * v * v * v);
            v = 0.5f * v * (1.0f + tanhf(u));
          } else if (epi == 2) {
            v = res[(size_t)rowg * ldres + colg] + gate[colg] * v;
          }
          C[(size_t)rowg * ldc + colg] = v;
        }
      }
    }
}

// ---------------------------------------------------------------------------
// Flash attention, f32 WMMA. One wave = one (16-query tile, head).
// Q pre-scaled by 1/sqrt(d). Streams 16-key tiles with online softmax.
// ---------------------------------------------------------------------------
__global__ __launch_bounds__(128) void attn_wmma_f32(
    const float* __restrict__ Q, const float* __restrict__ Kr,
    const float* __restrict__ QKV, float* __restrict__ O, int ldo, int Lq,
    int Lk) {
  __shared__ float Pl[4][16][BKP];  // 80B rows: aligned b64 reads
  int lane = threadIdx.x & 31, wave = threadIdx.x >> 5;
  int fr = lane & 15, hi = lane >> 4;
  int q0 = (blockIdx.x * 4 + wave) * 16;
  int h = blockIdx.y;

  // preload Q fragments for the whole head_dim (32 x K4 steps)
  int qrow = min(q0 + fr, Lq - 1);
  const float* qp = Q + (size_t)qrow * HID + h * HD;
  v2f qf[32];
#pragma unroll
  for (int kk = 0; kk < 32; ++kk) {
    float2 t = *(const float2*)(qp + kk * 4 + 2 * hi);
    qf[kk].x = t.x;
    qf[kk].y = t.y;
  }

  v8f o[8];
#pragma unroll
  for (int i = 0; i < 8; ++i) o[i] = (v8f){};
  float rmax[8], rsum[8];
#pragma unroll
  for (int r = 0; r < 8; ++r) { rmax[r] = -3.0e38f; rsum[r] = 0.f; }

  int nkt = (Lk + 15) >> 4;
  for (int kt = 0; kt < nkt; ++kt) {
    int kb = kt * 16;
    // S = Q K^T   (16 x 16, K=128)
    v8f s = (v8f){};
    int krow = min(kb + fr, Lk - 1);
    const float* kp = Kr + (size_t)krow * HID + h * HD;
#pragma unroll
    for (int kk = 0; kk < 32; ++kk) {
      float2 t = *(const float2*)(kp + kk * 4 + 2 * hi);
      v2f bf;
      bf.x = t.x;
      bf.y = t.y;
      s = wmma4(qf[kk], bf, s);
    }
    float keyok = (kb + fr < Lk) ? 0.0f : -3.0e38f;
    float alpha[8];
#pragma unroll
    for (int r = 0; r < 8; ++r) {
      float sv = s[r] + keyok;
      float mx = sv;
      mx = fmaxf(mx, __shfl_xor(mx, 1));
      mx = fmaxf(mx, __shfl_xor(mx, 2));
      mx = fmaxf(mx, __shfl_xor(mx, 4));
      mx = fmaxf(mx, __shfl_xor(mx, 8));
      float nm = fmaxf(rmax[r], mx);
      float al = __expf(rmax[r] - nm);
      float p = __expf(sv - nm);
      float ps = p;
      ps += __shfl_xor(ps, 1);
      ps += __shfl_xor(ps, 2);
      ps += __shfl_xor(ps, 4);
      ps += __shfl_xor(ps, 8);
      rsum[r] = rsum[r] * al + ps;
      rmax[r] = nm;
      alpha[r] = al;
      Pl[wave][r + hi * 8][fr] = p;  // C-layout -> LDS, reread as A-layout
    }
#pragma unroll
    for (int nt = 0; nt < 8; ++nt)
#pragma unroll
      for (int r = 0; r < 8; ++r) o[nt][r] *= alpha[r];
    __syncthreads();
    // O += P V   (16 x 128, K=16)
    const float* vb = QKV + 2 * (size_t)HID + h * HD;
#pragma unroll
    for (int kk = 0; kk < 16; kk += 4) {
      int pc = kk + 2 * hi;
      float2 tp = *(const float2*)(&Pl[wave][fr][pc]);
      v2f pa;
      pa.x = tp.x;
      pa.y = tp.y;
      int vr0 = min(kb + pc, Lk - 1);
      int vr1 = min(kb + pc + 1, Lk - 1);
#pragma unroll
      for (int nt = 0; nt < 8; ++nt) {
        v2f bv;
        bv.x = vb[(size_t)vr0 * QKVW + nt * 16 + fr];
        bv.y = vb[(size_t)vr1 * QKVW + nt * 16 + fr];
        o[nt] = wmma4(pa, bv, o[nt]);
      }
    }
    __syncthreads();
  }
  // normalize + store
#pragma unroll
  for (int nt = 0; nt < 8; ++nt) {
    int col = h * HD + nt * 16 + fr;
#pragma unroll
    for (int r = 0; r < 8; ++r) {
      int row = q0 + r + hi * 8;
      if (row < Lq) O[(size_t)row * ldo + col] = o[nt][r] * __frcp_rn(rsum[r]);
    }
  }
}

// ---------------------------------------------------------------------------
extern "C" void kernel_launch(void* const* d_in, const int* in_sizes, int n_in,
                              void* d_out, int out_size, void* d_ws,
                              size_t ws_size, hipStream_t stream) {
  const float* x = (const float*)d_in[0];
  const float* concepts = (const float*)d_in[1];
  const float* vec = (const float*)d_in[2];
  const float* pe = (const float*)d_in[3];
  const float* cpe = (const float*)d_in[4];
  const float* w1 = (const float*)d_in[5];
  const float* b1 = (const float*)d_in[6];
  const float* w2 = (const float*)d_in[7];
  const float* b2 = (const float*)d_in[8];
  const float* qsc = (const float*)d_in[9];
  const float* ksc = (const float*)d_in[10];
  const float* mw = (const float*)d_in[11];
  const float* mb = (const float*)d_in[12];
  float* out = (float*)d_out;

  float* wsf = (float*)d_ws;
  float* sv = wsf;                            // 1536
  float* m = sv + HID;                        // 4608 (shift|scale|gate)
  float* xm = m + 3 * HID;                    // LX*1536
  float* qkv = xm + (size_t)LXN * HID;        // LX*4608
  float* qr = qkv + (size_t)LXN * QKVW;       // LX*1536
  float* kr = qr + (size_t)LXN * HID;         // LX*1536
  float* a2 = kr + (size_t)LXN * HID;         // LX*7680  (attn | gelu(mlp))
  float* gate = m + 2 * HID;

  // modulation
  silu_k<<<(HID + 255) / 256, 256, 0, stream>>>(vec, sv);
  modm_k<<<(3 * HID + 255) / 256, 256, 0, stream>>>(sv, mw, mb, m);

  // ---------------- block 1 : full sequence (L = 4352) ----------------
  ln_mod_k<<<LXN, 256, 0, stream>>>(x, LXN, x, m, xm);
  // qkv = xm @ w1[:4608]^T + b1
  gemm_wmma_f32<<<dim3(QKVW / BN, LXN / BM), 256, 0, stream>>>(
      xm, HID, w1, HID, b1, qkv, QKVW, LXN, QKVW, HID, 0, nullptr, 0, nullptr);
  // gelu(mlp) -> a2[:,1536:]
  gemm_wmma_f32<<<dim3(MLPW / BN, LXN / BM), 256, 0, stream>>>(
      xm, HID, w1 + (size_t)QKVW * HID, HID, b1 + QKVW, a2 + HID, CATW, LXN,
      MLPW, HID, 1, nullptr, 0, nullptr);
  rope_rms_k<<<(LXN * HEADS) / 8, 256, 0, stream>>>(qkv, pe, qsc, ksc, qr, kr,
                                                    LXN);
  attn_wmma_f32<<<dim3(LXN / 64, HEADS), 128, 0, stream>>>(qr, kr, qkv, a2,
                                                           CATW, LXN, LXN);
  // out_x = x + gate * (a2 @ w2^T + b2)
  gemm_wmma_f32<<<dim3(HID / BN, LXN / BM), 256, 0, stream>>>(
      a2, CATW, w2, CATW, b2, out, HID, LXN, HID, CATW, 2, x, HID, gate);

  // ------------- block 2 : concepts + image (L = 4116, 20 live rows) -------
  ln_mod_k<<<LCN, 256, 0, stream>>>(concepts, NCON, x + (size_t)256 * HID, m,
                                    xm);
  // full K/V needed -> qkv for all rows
  gemm_wmma_f32<<<dim3(QKVW / BN, (LCN + BM - 1) / BM), 256, 0, stream>>>(
      xm, HID, w1, HID, b1, qkv, QKVW, LCN, QKVW, HID, 0, nullptr, 0, nullptr);
  // MLP only matters for the 20 concept rows
  gemm_wmma_f32<<<dim3(MLPW / BN, 1), 256, 0, stream>>>(
      xm, HID, w1 + (size_t)QKVW * HID, HID, b1 + QKVW, a2 + HID, CATW, NCON,
      MLPW, HID, 1, nullptr, 0, nullptr);
  rope_rms_k<<<(LCN * HEADS) / 8, 256, 0, stream>>>(qkv, cpe, qsc, ksc, qr, kr,
                                                    LCN);
  // only 20 queries survive (2 q-tiles), all 4116 keys
  attn_wmma_f32<<<dim3(1, HEADS), 128, 0, stream>>>(qr, kr, qkv, a2, CATW,
                                                    NCON, LCN);
  gemm_wmma_f32<<<dim3(HID / BN, 1), 256, 0, stream>>>(
      a2, CATW, w2, CATW, b2, out + (size_t)LXN * HID, HID, NCON, HID, CATW, 2,
      concepts, HID, gate);
}